// BertMentionDetectorModel_78864189489651
// MI455X (gfx1250) — compile-verified
//
#include <hip/hip_runtime.h>
#include <hip/hip_bf16.h>

// Problem constants (from reference setup_inputs)
#define BB   8
#define LL   1024
#define HH   768
#define MM   16
#define BIGF 1e31f
#define MAXLEN 256

typedef float v2f __attribute__((ext_vector_type(2)));
typedef float v8f __attribute__((ext_vector_type(8)));

// ---------------------------------------------------------------------------
// Kernel 1: logits = hs @ W + b via V_WMMA_F32_16X16X4_F32, then apply mask,
// scatter into ms/me/mm arrays [B*L].
// 64 blocks x 256 threads (8 waves). Each wave owns one 16-token tile.
// W is zero-padded 3 -> 16 columns and transposed into LDS once per block so
// the inner loop is branch-free: global_load_b64 (A) + ds_load_b64 (B) + wmma.
// ---------------------------------------------------------------------------
__global__ __launch_bounds__(256)
void md_logits_wmma(const float* __restrict__ hs,
                    const int*   __restrict__ masks,
                    const float* __restrict__ W,
                    const float* __restrict__ bias,
                    float* __restrict__ msArr,
                    float* __restrict__ meArr,
                    float* __restrict__ mmArr) {
    // Wt[n][k]: transposed, zero-padded weights (16 x 768 f32 = 48 KB)
    __shared__ float Wt[16 * HH];

    for (int idx = threadIdx.x; idx < 16 * HH; idx += 256) {
        const int n = idx / HH;
        const int k = idx - n * HH;
        Wt[idx] = (n < 3) ? W[k * 3 + n] : 0.0f;
    }
    __syncthreads();

    const int wave = threadIdx.x >> 5;        // 0..7
    const int lane = threadIdx.x & 31;        // 0..31
    const int tile = blockIdx.x * 8 + wave;   // 0 .. (B*L/16 - 1)
    const int base = tile * 16;               // first token (row of [B*L, H])

    const int m  = lane & 15;                 // A row (== B col n)
    const int kh = lane >> 4;                 // K-half select (0/1)

    const float* aRow = hs + (size_t)(base + m) * HH + 2 * kh;
    const float* bRow = Wt + m * HH + 2 * kh;

    v8f c = {};
    for (int k = 0; k < HH; k += 4) {
        // A: 16x4 f32 tile of hidden_states rows (one 8-byte global load)
        const v2f a = *(const v2f*)(aRow + k);
        // B: 4x16 f32 tile of padded W (one 8-byte LDS load)
        const v2f bv = *(const v2f*)(bRow + k);
        c = __builtin_amdgcn_wmma_f32_16x16x4_f32(
                /*neg_a=*/false, a, /*neg_b=*/false, bv,
                /*c_mod=*/(short)0, c, /*reuse_a=*/false, /*reuse_b=*/false);
    }

    // C/D layout: lane -> col n = lane&15; VGPR r -> row m = 8*(lane>>4)+r
    const int n = lane & 15;
    if (n < 3) {
        const float bn = bias[n];
        #pragma unroll
        for (int r = 0; r < 8; ++r) {
            const int row   = 8 * kh + r;
            const int token = base + row;
            const float mv  = (masks[token] == 0) ? -BIGF : 1.0f;
            const float val = (c[r] + bn) + mv;
            if (n == 0)      msArr[token] = val;
            else if (n == 1) meArr[token] = val;
            else             mmArr[token] = val;
        }
    }
}

// ---------------------------------------------------------------------------
// Kernel 2: inclusive cumsum of mm along L (one workgroup of 1024 per batch),
// Hillis-Steele scan in LDS. Also zeroes the loss accumulators.
// ---------------------------------------------------------------------------
__global__ __launch_bounds__(LL)
void md_cumsum(const float* __restrict__ mmArr,
               float* __restrict__ cumArr,
               float* __restrict__ accum) {
    __shared__ float buf[2][LL];
    const int b = blockIdx.x;
    const int t = threadIdx.x;
    if (b == 0 && t < 2) accum[t] = 0.0f;

    buf[0][t] = mmArr[b * LL + t];
    __syncthreads();
    int src = 0;
    for (int off = 1; off < LL; off <<= 1) {
        float x = buf[src][t];
        if (t >= off) x += buf[src][t - off];
        buf[1 - src][t] = x;
        src = 1 - src;
        __syncthreads();
    }
    cumArr[b * LL + t] = buf[src][t];
}

// ---------------------------------------------------------------------------
// Kernel 3: scores[b,i,j] for 16 rows per workgroup; me/cum rows cached in
// LDS; accumulates softplus(x) over valid spans + valid count via block
// reduction + global float atomics.
// ---------------------------------------------------------------------------
__global__ __launch_bounds__(256)
void md_scores(const float* __restrict__ msArr,
               const float* __restrict__ meArr,
               const float* __restrict__ mmArr,
               const float* __restrict__ cumArr,
               float* __restrict__ out,
               float* __restrict__ accum) {
    __shared__ float sme[LL];
    __shared__ float scum[LL];
    __shared__ float r1[256];
    __shared__ float r2[256];

    const int b = blockIdx.y;
    const int t = threadIdx.x;

    for (int idx = t; idx < LL; idx += 256) {
        sme[idx]  = meArr[b * LL + idx];
        scum[idx] = cumArr[b * LL + idx];
    }
    __syncthreads();

    const int i    = blockIdx.x * 16 + (t >> 4);
    const float msi  = msArr[b * LL + i];
    const float cumi = cumArr[b * LL + i];
    const float mmi  = mmArr[b * LL + i];

    float lsum = 0.0f;
    float lcnt = 0.0f;
    const size_t rowbase = ((size_t)(b * LL + i)) * LL;

    #pragma unroll 4
    for (int jj = 0; jj < LL / 16; ++jj) {
        const int j = (t & 15) + (jj << 4);
        float x;
        if (i <= j) {
            const float span = (scum[j] - cumi) + mmi;   // match ref association
            x = (msi + sme[j]) + span;
        } else {
            x = -BIGF;
        }
        out[rowbase + j] = x;
        const bool valid = (i <= j) && (x > -BIGF) && ((j - i) < MAXLEN);
        if (valid) {
            const float ax = fabsf(x);
            lsum += fmaxf(x, 0.0f) + log1pf(__expf(-ax));
            lcnt += 1.0f;
        }
    }

    r1[t] = lsum;
    r2[t] = lcnt;
    __syncthreads();
    for (int s = 128; s > 0; s >>= 1) {
        if (t < s) { r1[t] += r1[t + s]; r2[t] += r2[t + s]; }
        __syncthreads();
    }
    if (t == 0) {
        atomicAdd(&accum[0], r1[0]);
        atomicAdd(&accum[1], r2[0]);
    }
}

// ---------------------------------------------------------------------------
// Kernel 4: gold-target correction: per = ... - x*target. targets built by
// scatter-max, so deduplicate (s,e) within a batch, then subtract x for each
// distinct valid gold span that is also a 'valid' span.
// ---------------------------------------------------------------------------
__global__ __launch_bounds__(BB * MM)
void md_gold(const int* __restrict__ starts,
             const int* __restrict__ ends,
             const float* __restrict__ msArr,
             const float* __restrict__ meArr,
             const float* __restrict__ mmArr,
             const float* __restrict__ cumArr,
             float* __restrict__ accum) {
    const int t = threadIdx.x;
    const int b = t >> 4;
    const int m = t & 15;

    const int s0 = starts[b * MM + m];
    const int e0 = ends[b * MM + m];
    const bool gv = (!((s0 == 0) && (e0 == 0))) && (s0 != -1);
    if (!gv) return;

    // dedup: only the first occurrence of (s0,e0) among valid golds counts
    for (int m2 = 0; m2 < m; ++m2) {
        const int s2 = starts[b * MM + m2];
        const int e2 = ends[b * MM + m2];
        const bool gv2 = (!((s2 == 0) && (e2 == 0))) && (s2 != -1);
        if (gv2 && s2 == s0 && e2 == e0) return;
    }

    const int i = s0, j = e0;
    if (i < 0 || j < 0 || i >= LL || j >= LL) return;
    if (!(i <= j) || (j - i) >= MAXLEN) return;

    const float span = (cumArr[b * LL + j] - cumArr[b * LL + i]) + mmArr[b * LL + i];
    const float x    = (msArr[b * LL + i] + meArr[b * LL + j]) + span;
    if (x > -BIGF) atomicAdd(&accum[0], -x);
}

// ---------------------------------------------------------------------------
// Kernel 5: finalize loss scalar at d_out[B*L*L]
// ---------------------------------------------------------------------------
__global__ void md_finalize(const float* __restrict__ accum,
                            float* __restrict__ out) {
    out[(size_t)BB * LL * LL] = accum[0] / accum[1];
}

// ---------------------------------------------------------------------------
extern "C" void kernel_launch(void* const* d_in, const int* in_sizes, int n_in,
                              void* d_out, int out_size, void* d_ws, size_t ws_size,
                              hipStream_t stream) {
    const float* hs     = (const float*)d_in[0];   // [B,L,H]
    const int*   masks  = (const int*)d_in[1];     // [B,L]
    const int*   starts = (const int*)d_in[2];     // [B,M]
    const int*   ends   = (const int*)d_in[3];     // [B,M]
    const float* W      = (const float*)d_in[4];   // [H,3]
    const float* bias   = (const float*)d_in[5];   // [3]
    float* out = (float*)d_out;                    // [B,L,L] scores + 1 loss

    // workspace layout (floats)
    float* ws     = (float*)d_ws;
    float* msArr  = ws;                 // B*L
    float* meArr  = ws + BB * LL;       // B*L
    float* mmArr  = ws + 2 * BB * LL;   // B*L
    float* cumArr = ws + 3 * BB * LL;   // B*L
    float* accum  = ws + 4 * BB * LL;   // [num, cnt]

    // 1) projection via WMMA + mask  (512 tiles, 8 waves/block)
    md_logits_wmma<<<dim3((BB * LL) / (16 * 8)), dim3(256), 0, stream>>>(
        hs, masks, W, bias, msArr, meArr, mmArr);

    // 2) cumsum per batch + zero accumulators
    md_cumsum<<<dim3(BB), dim3(LL), 0, stream>>>(mmArr, cumArr, accum);

    // 3) scores + loss partials
    md_scores<<<dim3(LL / 16, BB), dim3(256), 0, stream>>>(
        msArr, meArr, mmArr, cumArr, out, accum);

    // 4) gold-target correction
    md_gold<<<dim3(1), dim3(BB * MM), 0, stream>>>(
        starts, ends, msArr, meArr, mmArr, cumArr, accum);

    // 5) finalize scalar loss
    md_finalize<<<dim3(1), dim3(1), 0, stream>>>(accum, out);
}